// AtomicGNN_21191368638509
// MI455X (gfx1250) — compile-verified
//
#include <hip/hip_runtime.h>
#include <hip/hip_bf16.h>

typedef __attribute__((ext_vector_type(2))) float v2f;
typedef __attribute__((ext_vector_type(8))) float v8f;

#define NEG_SLOPE 0.2f

__device__ __forceinline__ unsigned ordEnc(float f) {
  unsigned u = __float_as_uint(f);
  return (u & 0x80000000u) ? ~u : (u | 0x80000000u);
}
__device__ __forceinline__ float ordDec(unsigned u) {
  return (u & 0x80000000u) ? __uint_as_float(u ^ 0x80000000u)
                           : __uint_as_float(~u);
}

// ---------------- generic fill ----------------
__global__ void gat_fill_u32(unsigned* __restrict__ p, unsigned v, long long n) {
  long long i = (long long)blockIdx.x * blockDim.x + threadIdx.x;
  if (i < n) p[i] = v;
}

// ---------------- self-loop attr: per-dst count + sum ----------------
__global__ void gat_count_loop(const int* __restrict__ dst,
                               const float* __restrict__ ea,
                               float* __restrict__ cnt,
                               float* __restrict__ la, int E) {
  long long i = (long long)blockIdx.x * blockDim.x + threadIdx.x;
  if (i >= (long long)E * 16) return;
  int e = (int)(i >> 4), j = (int)(i & 15);
  int d = dst[e];
  atomicAdd(&la[(long long)d * 16 + j], ea[i]);
  if (j == 0) atomicAdd(&cnt[d], 1.0f);
}

__global__ void gat_loop_div(const float* __restrict__ cnt,
                             float* __restrict__ la, int Nn) {
  long long i = (long long)blockIdx.x * blockDim.x + threadIdx.x;
  if (i >= (long long)Nn * 16) return;
  float c = cnt[i >> 4];
  c = (c < 1.0f) ? 1.0f : c;
  la[i] = la[i] / c;
}

// ---------------- M_edge[k][h] = sum_c W_edge[k, h*32+c] * att_edge[h,c] ----------------
__global__ void gat_medge(const float* __restrict__ We,
                          const float* __restrict__ ae,
                          float* __restrict__ medge) {
  int t = threadIdx.x;            // 64 threads
  int k = t >> 2, h = t & 3;
  float s = 0.0f;
  for (int c = 0; c < 32; ++c)
    s += We[k * 128 + h * 32 + c] * ae[h * 32 + c];
  medge[k * 4 + h] = s;
}

// ---------------- WMMA GEMM: Y[N,128] = X[N,128] @ W[128,128] (f32) ----------------
// One wave computes a 16-row x 128-col strip: 8 accumulators, K-loop of 32 x K=4.
__global__ __launch_bounds__(256) void gat_gemm_wmma(
    const float* __restrict__ X, const float* __restrict__ W,
    float* __restrict__ Y, int Nrows) {
  int wave = (int)(((long long)blockIdx.x * blockDim.x + threadIdx.x) >> 5);
  int lane = threadIdx.x & 31;
  int m0 = wave * 16;
  if (m0 >= Nrows) return;

  v8f acc[8] = {};
  int half = lane >> 4;           // 0: K pair 0/1,  1: K pair 2/3
  int l16 = lane & 15;
  int arow = m0 + l16;
  if (arow >= Nrows) arow = Nrows - 1;   // clamp loads, guard stores below
  const float* xr = X + (long long)arow * 128;

  for (int k0 = 0; k0 < 128; k0 += 4) {
    v2f a;
    a.x = xr[k0 + half * 2];
    a.y = xr[k0 + half * 2 + 1];
#pragma unroll
    for (int nt = 0; nt < 8; ++nt) {
      int col = nt * 16 + l16;
      v2f b;
      b.x = W[(k0 + half * 2) * 128 + col];
      b.y = W[(k0 + half * 2 + 1) * 128 + col];
      acc[nt] = __builtin_amdgcn_wmma_f32_16x16x4_f32(
          false, a, false, b, (short)0, acc[nt], false, false);
    }
  }
#pragma unroll
  for (int nt = 0; nt < 8; ++nt) {
#pragma unroll
    for (int r = 0; r < 8; ++r) {
      int row = m0 + r + half * 8;
      if (row < Nrows) Y[(long long)row * 128 + nt * 16 + l16] = acc[nt][r];
    }
  }
}

// ---------------- a_src/a_dst: per-node per-head dot (wave = one head) ----------------
__global__ __launch_bounds__(128) void gat_attdot(
    const float* __restrict__ xh, const float* __restrict__ as,
    const float* __restrict__ ad, float* __restrict__ a_src,
    float* __restrict__ a_dst, int Nn) {
  int n = blockIdx.x;
  int t = threadIdx.x;            // t = h*32 + c
  float v = xh[(long long)n * 128 + t];
  float ps = v * as[t];
  float pd = v * ad[t];
  for (int o = 16; o > 0; o >>= 1) {
    ps += __shfl_down(ps, o, 32);
    pd += __shfl_down(pd, o, 32);
  }
  if ((t & 31) == 0) {
    int h = t >> 5;
    a_src[n * 4 + h] = ps;
    a_dst[n * 4 + h] = pd;
  }
}

// ---------------- alpha = leaky_relu(a_src[s]+a_dst[d]+a_edge) ; segment max ----------------
__global__ void gat_alpha(const int* __restrict__ srcI,
                          const int* __restrict__ dstI,
                          const float* __restrict__ ea,
                          const float* __restrict__ la,
                          const float* __restrict__ a_src,
                          const float* __restrict__ a_dst,
                          const float* __restrict__ medge,
                          float* __restrict__ alpha,
                          unsigned* __restrict__ mseg, int E, int Nn) {
  long long f = (long long)blockIdx.x * blockDim.x + threadIdx.x;
  long long F = (long long)E + Nn;
  if (f >= F) return;
  int s, d;
  const float* row;
  if (f < (long long)E) {
    s = srcI[f]; d = dstI[f]; row = ea + f * 16;
  } else {
    s = (int)(f - E); d = s; row = la + (long long)s * 16;
  }
  float e0[16];
#pragma unroll
  for (int k = 0; k < 16; ++k) e0[k] = row[k];
#pragma unroll
  for (int h = 0; h < 4; ++h) {
    float aE = 0.0f;
#pragma unroll
    for (int k = 0; k < 16; ++k) aE += e0[k] * medge[k * 4 + h];
    float v = a_src[(long long)s * 4 + h] + a_dst[(long long)d * 4 + h] + aE;
    v = (v > 0.0f) ? v : NEG_SLOPE * v;
    alpha[f * 4 + h] = v;
    atomicMax(&mseg[(long long)d * 4 + h], ordEnc(v));
  }
}

// ---------------- p = exp(alpha - m[dst]) ; segment sum ----------------
__global__ void gat_expsum(const int* __restrict__ dstI,
                           float* __restrict__ alpha,
                           const unsigned* __restrict__ mseg,
                           float* __restrict__ sseg, int E, int Nn) {
  long long i = (long long)blockIdx.x * blockDim.x + threadIdx.x;
  long long F4 = ((long long)E + Nn) * 4;
  if (i >= F4) return;
  long long f = i >> 2;
  int h = (int)(i & 3);
  int d = (f < (long long)E) ? dstI[f] : (int)(f - E);
  float mv = ordDec(mseg[(long long)d * 4 + h]);
  if (!(mv > -__builtin_inff() && mv < __builtin_inff())) mv = 0.0f;
  float p = __expf(alpha[i] - mv);
  alpha[i] = p;
  atomicAdd(&sseg[(long long)d * 4 + h], p);
}

// ---------------- out[dst] += (p/s) * xh[src] : wave per edge, float4 per lane ----------------
__global__ __launch_bounds__(256) void gat_aggr(
    const int* __restrict__ srcI, const int* __restrict__ dstI,
    const float* __restrict__ alpha, const float* __restrict__ sseg,
    const float* __restrict__ xh, float* __restrict__ out, int E, int Nn) {
  long long wid = ((long long)blockIdx.x * blockDim.x + threadIdx.x) >> 5;
  int lane = threadIdx.x & 31;
  long long F = (long long)E + Nn;
  if (wid >= F) return;
  int s, d;
  if (wid < (long long)E) { s = srcI[wid]; d = dstI[wid]; }
  else { s = (int)(wid - E); d = s; }
  int c0 = lane * 4;
  int h = lane >> 3;              // 32 channels per head -> 8 lanes per head
  float a = alpha[wid * 4 + h] / (sseg[(long long)d * 4 + h] + 1e-16f);
  const float4 xv = *(const float4*)(xh + (long long)s * 128 + c0);
  float* o = out + (long long)d * 128 + c0;
  atomicAdd(o + 0, a * xv.x);
  atomicAdd(o + 1, a * xv.y);
  atomicAdd(o + 2, a * xv.z);
  atomicAdd(o + 3, a * xv.w);
}

// ---------------- epilogue: y = (relu?)(y + bias[c]) ----------------
__global__ void gat_bias_act(float* __restrict__ y,
                             const float* __restrict__ bias, long long n,
                             int relu) {
  long long i = (long long)blockIdx.x * blockDim.x + threadIdx.x;
  if (i >= n) return;
  float v = y[i] + bias[i & 127];
  if (relu) v = (v > 0.0f) ? v : 0.0f;
  y[i] = v;
}

// ================= host =================
static void fillN(unsigned* p, unsigned v, long long n, hipStream_t s) {
  if (n <= 0) return;
  long long b = (n + 255) / 256;
  hipLaunchKernelGGL(gat_fill_u32, dim3((unsigned)b), dim3(256), 0, s, p, v, n);
}

extern "C" void kernel_launch(void* const* d_in, const int* in_sizes, int n_in,
                              void* d_out, int out_size, void* d_ws,
                              size_t ws_size, hipStream_t stream) {
  const int N = in_sizes[0] / 128;
  const int E = in_sizes[1] / 2;

  const float* x  = (const float*)d_in[0];
  const int* ei   = (const int*)d_in[1];
  const int* srcI = ei;
  const int* dstI = ei + E;
  const float* ea = (const float*)d_in[2];

  const float* Wl[2]  = {(const float*)d_in[3], (const float*)d_in[9]};
  const float* asl[2] = {(const float*)d_in[4], (const float*)d_in[10]};
  const float* adl[2] = {(const float*)d_in[5], (const float*)d_in[11]};
  const float* Wel[2] = {(const float*)d_in[6], (const float*)d_in[12]};
  const float* ael[2] = {(const float*)d_in[7], (const float*)d_in[13]};
  const float* bl[2]  = {(const float*)d_in[8], (const float*)d_in[14]};

  // workspace layout (floats)
  float* wsf = (float*)d_ws;
  size_t off = 0;
  float* cnt   = wsf + off; off += (size_t)N;
  float* la    = wsf + off; off += (size_t)N * 16;
  float* medge = wsf + off; off += 64;
  float* xh    = wsf + off; off += (size_t)N * 128;
  float* a_src = wsf + off; off += (size_t)N * 4;
  float* a_dst = wsf + off; off += (size_t)N * 4;
  float* alpha = wsf + off; off += ((size_t)E + N) * 4;
  unsigned* mseg = (unsigned*)(wsf + off); off += (size_t)N * 4;
  float* sseg  = wsf + off; off += (size_t)N * 4;
  float* h1    = wsf + off; off += (size_t)N * 128;
  (void)ws_size; (void)n_in; (void)out_size;

  const long long F = (long long)E + N;

  // self-loop mean edge attrs (layer independent)
  fillN((unsigned*)cnt, 0u, N, stream);
  fillN((unsigned*)la, 0u, (long long)N * 16, stream);
  {
    long long n = (long long)E * 16;
    hipLaunchKernelGGL(gat_count_loop, dim3((unsigned)((n + 255) / 256)),
                       dim3(256), 0, stream, dstI, ea, cnt, la, E);
    long long m = (long long)N * 16;
    hipLaunchKernelGGL(gat_loop_div, dim3((unsigned)((m + 255) / 256)),
                       dim3(256), 0, stream, cnt, la, N);
  }

  for (int l = 0; l < 2; ++l) {
    const float* xin = (l == 0) ? x : h1;
    float* outb = (l == 0) ? h1 : (float*)d_out;

    hipLaunchKernelGGL(gat_medge, dim3(1), dim3(64), 0, stream, Wel[l], ael[l],
                       medge);

    { // WMMA GEMM: one wave per 16 rows
      long long waves = ((long long)N + 15) / 16;
      long long blocks = (waves + 7) / 8;   // 8 waves / 256-thread block
      hipLaunchKernelGGL(gat_gemm_wmma, dim3((unsigned)blocks), dim3(256), 0,
                         stream, xin, Wl[l], xh, N);
    }

    hipLaunchKernelGGL(gat_attdot, dim3((unsigned)N), dim3(128), 0, stream, xh,
                       asl[l], adl[l], a_src, a_dst, N);

    fillN(mseg, 0x007fffffu, (long long)N * 4, stream);      // encode(-inf)
    fillN((unsigned*)sseg, 0u, (long long)N * 4, stream);
    fillN((unsigned*)outb, 0u, (long long)N * 128, stream);

    hipLaunchKernelGGL(gat_alpha, dim3((unsigned)((F + 255) / 256)), dim3(256),
                       0, stream, srcI, dstI, ea, la, a_src, a_dst, medge,
                       alpha, mseg, E, N);

    {
      long long n4 = F * 4;
      hipLaunchKernelGGL(gat_expsum, dim3((unsigned)((n4 + 255) / 256)),
                         dim3(256), 0, stream, dstI, alpha, mseg, sseg, E, N);
    }

    { // wave per edge
      long long blocks = (F + 7) / 8;
      hipLaunchKernelGGL(gat_aggr, dim3((unsigned)blocks), dim3(256), 0, stream,
                         srcI, dstI, alpha, sseg, xh, outb, E, N);
    }

    {
      long long n = (long long)N * 128;
      hipLaunchKernelGGL(gat_bias_act, dim3((unsigned)((n + 255) / 256)),
                         dim3(256), 0, stream, outb, bl[l], n, (l == 0) ? 1 : 0);
    }
  }
}